// GNNModel_19121194402005
// MI455X (gfx1250) — compile-verified
//
#include <hip/hip_runtime.h>

typedef float v2f __attribute__((ext_vector_type(2)));
typedef float v8f __attribute__((ext_vector_type(8)));

#define D_FEAT 64

// ---------------------------------------------------------------- utilities
__global__ void zero_f32(float* __restrict__ p, long n) {
    long i = (long)blockIdx.x * blockDim.x + threadIdx.x;
    long stride = (long)gridDim.x * blockDim.x;
    for (; i < n; i += stride) p[i] = 0.0f;
}

// One thread per edge: accumulate out-degree (src) and in-degree (dst).
__global__ void degree_kernel(const int* __restrict__ src, const int* __restrict__ dst,
                              float* __restrict__ deg_src, float* __restrict__ deg_dst, int E) {
    int e = blockIdx.x * blockDim.x + threadIdx.x;
    if (e < E) {
        atomicAdd(&deg_src[src[e]], 1.0f);
        atomicAdd(&deg_dst[dst[e]], 1.0f);
    }
}

// deg -> rsqrt(max(deg, 1)) in place for both arrays.
__global__ void norm_kernel(float* __restrict__ deg_src, float* __restrict__ deg_dst, int N) {
    int i = blockIdx.x * blockDim.x + threadIdx.x;
    if (i < N) {
        deg_src[i] = rsqrtf(fmaxf(deg_src[i], 1.0f));
        deg_dst[i] = rsqrtf(fmaxf(deg_dst[i], 1.0f));
    }
}

// SpMM scatter: one wave (32 lanes) per edge; each lane moves a float2
// (cols 2*lane, 2*lane+1). norm_src is folded in at scatter time.
__global__ void spmm_kernel(const float* __restrict__ x, const float* __restrict__ norm_src,
                            const int* __restrict__ src, const int* __restrict__ dst,
                            float* __restrict__ msg, int E) {
    long gid  = (long)blockIdx.x * blockDim.x + threadIdx.x;
    long e    = gid >> 5;
    int  lane = (int)(gid & 31);
    if (e >= E) return;
    int s = src[e];
    int d = dst[e];
    float ns = norm_src[s];
    float2 v = ((const float2*)(x + (long)s * D_FEAT))[lane];
    float* o = msg + (long)d * D_FEAT + lane * 2;
    atomicAdd(o + 0, v.x * ns);
    atomicAdd(o + 1, v.y * ns);
}

// Fused: OUT[N x 64] = ReLU( (msg * norm_dst[:,None]) @ W + b )
// One wave computes a 16-row x 64-col tile via V_WMMA_F32_16X16X4_F32.
// A fragment (16x4 f32): lanes 0-15 hold K=0,1; lanes 16-31 hold K=2,3.
// B fragment (4x16 f32): same split over K, N = lane&15.
// C/D (16x16 f32, v8f):  VGPR i -> M = i + 8*(lane>=16), N = lane&15.
__global__ void gemm_relu_wmma(const float* __restrict__ S, const float* __restrict__ norm_dst,
                               const float* __restrict__ W, const float* __restrict__ bias,
                               float* __restrict__ out, int N) {
    int lane   = threadIdx.x & 31;
    int wave   = threadIdx.x >> 5;
    int tile   = blockIdx.x * (blockDim.x >> 5) + wave;
    int ntiles = (N + 15) >> 4;
    if (tile >= ntiles) return;          // wave-uniform: EXEC stays all-1s for WMMA

    int row0 = tile << 4;
    int half = lane >> 4;                // 0: K pair {0,1}; 1: K pair {2,3}
    int l15  = lane & 15;

    int arow = row0 + l15;
    if (arow >= N) arow = N - 1;         // duplicate row load for ragged tail (N%16==0 here)
    float nd = norm_dst[arow];
    const float* Srow = S + (long)arow * D_FEAT + 2 * half;

    v8f c0 = {}, c1 = {}, c2 = {}, c3 = {};

    for (int k = 0; k < D_FEAT; k += 4) {
        v2f a;
        a.x = Srow[k]     * nd;
        a.y = Srow[k + 1] * nd;

        const float* Wk = W + (long)(k + 2 * half) * D_FEAT + l15;
        v2f b0, b1, b2, b3;
        b0.x = Wk[0];  b0.y = Wk[D_FEAT + 0];
        b1.x = Wk[16]; b1.y = Wk[D_FEAT + 16];
        b2.x = Wk[32]; b2.y = Wk[D_FEAT + 32];
        b3.x = Wk[48]; b3.y = Wk[D_FEAT + 48];

        c0 = __builtin_amdgcn_wmma_f32_16x16x4_f32(false, a, false, b0, (short)0, c0, false, false);
        c1 = __builtin_amdgcn_wmma_f32_16x16x4_f32(false, a, false, b1, (short)0, c1, false, false);
        c2 = __builtin_amdgcn_wmma_f32_16x16x4_f32(false, a, false, b2, (short)0, c2, false, false);
        c3 = __builtin_amdgcn_wmma_f32_16x16x4_f32(false, a, false, b3, (short)0, c3, false, false);
    }

    float bb0 = bias[l15], bb1 = bias[16 + l15], bb2 = bias[32 + l15], bb3 = bias[48 + l15];
#pragma unroll
    for (int i = 0; i < 8; ++i) {
        int row = row0 + i + 8 * half;
        if (row < N) {
            float* o = out + (long)row * D_FEAT + l15;
            o[0]  = fmaxf(c0[i] + bb0, 0.0f);
            o[16] = fmaxf(c1[i] + bb1, 0.0f);
            o[32] = fmaxf(c2[i] + bb2, 0.0f);
            o[48] = fmaxf(c3[i] + bb3, 0.0f);
        }
    }
}

extern "C" void kernel_launch(void* const* d_in, const int* in_sizes, int n_in,
                              void* d_out, int out_size, void* d_ws, size_t ws_size,
                              hipStream_t stream) {
    const float* x   = (const float*)d_in[0];
    const int*   src = (const int*)  d_in[1];
    const int*   dst = (const int*)  d_in[2];
    const float* W1  = (const float*)d_in[3];
    const float* b1  = (const float*)d_in[4];
    const float* W2  = (const float*)d_in[5];
    const float* b2  = (const float*)d_in[6];
    float* out = (float*)d_out;

    const int N = in_sizes[0] / D_FEAT;   // 100000
    const int E = in_sizes[1];            // 1600000

    // Workspace carve (floats): [norm_src N][norm_dst N][msg N*64][h1 N*64] ~= 52 MB
    float* ws       = (float*)d_ws;
    float* norm_src = ws;
    float* norm_dst = ws + N;
    float* msg      = ws + 2L * N;
    float* h1       = msg + (long)N * D_FEAT;

    const int ZB = 2048;                   // grid-stride zero blocks
    long msg_elems = (long)N * D_FEAT;

    // Degrees -> norms
    zero_f32<<<ZB, 256, 0, stream>>>(norm_src, 2L * N);
    degree_kernel<<<(E + 255) / 256, 256, 0, stream>>>(src, dst, norm_src, norm_dst, E);
    norm_kernel<<<(N + 255) / 256, 256, 0, stream>>>(norm_src, norm_dst, N);

    long spmm_threads = (long)E * 32;
    int  spmm_blocks  = (int)((spmm_threads + 255) / 256);
    int  tiles        = (N + 15) / 16;
    int  gemm_blocks  = (tiles + 7) / 8;   // 8 waves / 256-thread block

    // Layer 1
    zero_f32<<<ZB, 256, 0, stream>>>(msg, msg_elems);
    spmm_kernel<<<spmm_blocks, 256, 0, stream>>>(x, norm_src, src, dst, msg, E);
    gemm_relu_wmma<<<gemm_blocks, 256, 0, stream>>>(msg, norm_dst, W1, b1, h1, N);

    // Layer 2
    zero_f32<<<ZB, 256, 0, stream>>>(msg, msg_elems);
    spmm_kernel<<<spmm_blocks, 256, 0, stream>>>(h1, norm_src, src, dst, msg, E);
    gemm_relu_wmma<<<gemm_blocks, 256, 0, stream>>>(msg, norm_dst, W2, b2, out, N);
}